// GraphSAGE_Embs_2894807958004
// MI455X (gfx1250) — compile-verified
//
#include <hip/hip_runtime.h>

#define H 64

typedef __attribute__((ext_vector_type(2))) float v2f;
typedef __attribute__((ext_vector_type(8))) float v8f;

// ---------------------------------------------------------------------------
// Degree computation: deg_s[src]++ , deg_t[dst]++ over all binds edges.
// ---------------------------------------------------------------------------
__global__ __launch_bounds__(256)
void degree_kernel(const int* __restrict__ esrc, const int* __restrict__ edst,
                   float* __restrict__ deg_s, float* __restrict__ deg_t, int E)
{
    const int stride = gridDim.x * blockDim.x;
    for (int e = blockIdx.x * blockDim.x + threadIdx.x; e < E; e += stride) {
        atomicAdd(&deg_s[esrc[e]], 1.0f);
        atomicAdd(&deg_t[edst[e]], 1.0f);
    }
}

// ---------------------------------------------------------------------------
// Edge-parallel scatter-add: agg[scatter_idx[e], :] += x[gather_idx[e], :]
// 16 threads per edge, float4 (16B) per thread -> 256B per edge, coalesced.
// ---------------------------------------------------------------------------
__global__ __launch_bounds__(256)
void scatter_add_kernel(const float* __restrict__ xsrc,
                        const int* __restrict__ gather_idx,
                        const int* __restrict__ scatter_idx,
                        float* __restrict__ agg, int E)
{
    const long long nwork  = (long long)E * 16;
    const long long stride = (long long)gridDim.x * blockDim.x;
    for (long long t = (long long)blockIdx.x * blockDim.x + threadIdx.x;
         t < nwork; t += stride) {
        const int e = (int)(t >> 4);
        const int q = ((int)t & 15) << 2;          // feature quad
        const int s = gather_idx[e];
        const int d = scatter_idx[e];
        const float4 v = *reinterpret_cast<const float4*>(xsrc + (size_t)s * H + q);
        float* dst = agg + (size_t)d * H + q;
        atomicAdd(dst + 0, v.x);
        atomicAdd(dst + 1, v.y);
        atomicAdd(dst + 2, v.z);
        atomicAdd(dst + 3, v.w);
    }
}

// ---------------------------------------------------------------------------
// SAGE linear layer via full-precision f32 WMMA (V_WMMA_F32_16X16X4_F32):
//   out[n,:] = act( (agg[n,:]/max(deg[n],1)) @ Wl + xdst[n,:] @ Wr + b )
// One wave owns a 16x16 output tile and chains K=128 (64 from agg/Wl,
// 64 from xdst/Wr) through the f32 accumulator. Block = 8 waves = 32 rows.
//
// A-fragment (16x4 f32, v2f/lane): lanes 0-15 carry K=k,k+1 of row M=lane;
// lanes 16-31 carry K=k+2,k+3 of row M=lane-16.
// B-fragment (4x16 f32, v2f/lane): lanes 0-15 carry rows K=k,k+1 at N=lane;
// lanes 16-31 carry rows K=k+2,k+3 at N=lane-16.
// C/D (16x16 f32, v8f): lane<16 -> D[v][lane], lane>=16 -> D[8+v][lane-16].
// ---------------------------------------------------------------------------
template <bool RELU>
__global__ __launch_bounds__(256)
void sage_gemm_kernel(const float* __restrict__ agg, const float* __restrict__ deg,
                      const float* __restrict__ xdst,
                      const float* __restrict__ Wl, const float* __restrict__ Wr,
                      const float* __restrict__ bias,
                      float* __restrict__ out, int N)
{
    const int lane = threadIdx.x & 31;
    const int wave = threadIdx.x >> 5;
    const int col0 = (wave & 3) * 16;                    // output col tile
    const int row0 = blockIdx.x * 32 + (wave >> 2) * 16; // output row tile
    if (row0 >= N) return;                               // wave-uniform; EXEC stays full

    const int m    = lane & 15;
    const int kh   = (lane >> 4) << 1;                   // 0 or 2 (K half-select)
    const int mrow = row0 + m;
    const int nc   = col0 + m;

    const float inv = 1.0f / fmaxf(deg[mrow], 1.0f);
    const float* __restrict__ arow = agg  + (size_t)mrow * H;
    const float* __restrict__ xrow = xdst + (size_t)mrow * H;

    v8f c = {0.f, 0.f, 0.f, 0.f, 0.f, 0.f, 0.f, 0.f};

    // K = 0..63 : mean-aggregated neighbors through Wl
    #pragma unroll
    for (int k = 0; k < H; k += 4) {
        const int kk = k + kh;
        v2f a, b;
        a.x = arow[kk]     * inv;
        a.y = arow[kk + 1] * inv;
        b.x = Wl[kk * H + nc];
        b.y = Wl[(kk + 1) * H + nc];
        c = __builtin_amdgcn_wmma_f32_16x16x4_f32(false, a, false, b,
                                                  (short)0, c, false, false);
    }
    // K = 64..127 : self features through Wr
    #pragma unroll
    for (int k = 0; k < H; k += 4) {
        const int kk = k + kh;
        v2f a, b;
        a.x = xrow[kk];
        a.y = xrow[kk + 1];
        b.x = Wr[kk * H + nc];
        b.y = Wr[(kk + 1) * H + nc];
        c = __builtin_amdgcn_wmma_f32_16x16x4_f32(false, a, false, b,
                                                  (short)0, c, false, false);
    }

    const int rbase = row0 + ((lane >> 4) << 3);
    const float bv  = bias[nc];
    #pragma unroll
    for (int v = 0; v < 8; ++v) {
        float val = c[v] + bv;
        if (RELU) val = fmaxf(val, 0.0f);
        out[(size_t)(rbase + v) * H + nc] = val;
    }
}

// ---------------------------------------------------------------------------
// Link classifier: out[l] = dot(o_s[lsrc[l],:], o_t[ldst[l],:])
// One wave32 per label edge; float2 per lane + xor-shuffle reduction.
// ---------------------------------------------------------------------------
__global__ __launch_bounds__(256)
void dot_kernel(const float* __restrict__ os, const float* __restrict__ ot,
                const int* __restrict__ lsrc, const int* __restrict__ ldst,
                float* __restrict__ out, int EL)
{
    const int lane  = threadIdx.x & 31;
    const int wstep = (gridDim.x * blockDim.x) >> 5;
    for (int l = (blockIdx.x * blockDim.x + threadIdx.x) >> 5; l < EL; l += wstep) {
        const int s = lsrc[l];
        const int d = ldst[l];
        const float2 a = *reinterpret_cast<const float2*>(os + (size_t)s * H + lane * 2);
        const float2 b = *reinterpret_cast<const float2*>(ot + (size_t)d * H + lane * 2);
        float p = a.x * b.x + a.y * b.y;
        #pragma unroll
        for (int off = 16; off > 0; off >>= 1)
            p += __shfl_xor(p, off, 32);
        if (lane == 0) out[l] = p;
    }
}

// ---------------------------------------------------------------------------
extern "C" void kernel_launch(void* const* d_in, const int* in_sizes, int n_in,
                              void* d_out, int out_size, void* d_ws, size_t ws_size,
                              hipStream_t stream)
{
    const float* src_emb = (const float*)d_in[0];
    const float* tgt_emb = (const float*)d_in[1];
    const float* Wl_st1  = (const float*)d_in[2];
    const float* Wr_st1  = (const float*)d_in[3];
    const float* b_st1   = (const float*)d_in[4];
    const float* Wl_ts1  = (const float*)d_in[5];
    const float* Wr_ts1  = (const float*)d_in[6];
    const float* b_ts1   = (const float*)d_in[7];
    const float* Wl_st2  = (const float*)d_in[8];
    const float* Wr_st2  = (const float*)d_in[9];
    const float* b_st2   = (const float*)d_in[10];
    const float* Wl_ts2  = (const float*)d_in[11];
    const float* Wr_ts2  = (const float*)d_in[12];
    const float* b_ts2   = (const float*)d_in[13];
    const int*   ei      = (const int*)d_in[14];
    const int*   eli     = (const int*)d_in[15];

    const int NS = in_sizes[0] / H;
    const int NT = in_sizes[1] / H;
    const int E  = in_sizes[14] / 2;
    const int EL = in_sizes[15] / 2;
    const int NMAX = NS > NT ? NS : NT;
    (void)n_in; (void)out_size; (void)ws_size;

    const int* e_src = ei;
    const int* e_dst = ei + E;
    const int* l_src = eli;
    const int* l_dst = eli + EL;

    // Carve scratch out of d_ws (256B aligned slices).
    size_t off = 0;
    auto wsAlloc = [&](size_t bytes) -> void* {
        void* p = (char*)d_ws + off;
        off += (bytes + 255) & ~(size_t)255;
        return p;
    };
    float* deg_s = (float*)wsAlloc((size_t)NS * sizeof(float));
    float* deg_t = (float*)wsAlloc((size_t)NT * sizeof(float));
    float* agg   = (float*)wsAlloc((size_t)NMAX * H * sizeof(float));
    float* h_t   = (float*)wsAlloc((size_t)NT * H * sizeof(float));
    float* h_s   = (float*)wsAlloc((size_t)NS * H * sizeof(float));
    float* o_t   = (float*)wsAlloc((size_t)NT * H * sizeof(float));
    float* o_s   = (float*)wsAlloc((size_t)NS * H * sizeof(float));

    const int SCATTER_GRID = 4096;

    // Degrees (same edge index for both layers -> compute once).
    hipMemsetAsync(deg_s, 0, (size_t)NS * sizeof(float), stream);
    hipMemsetAsync(deg_t, 0, (size_t)NT * sizeof(float), stream);
    degree_kernel<<<1024, 256, 0, stream>>>(e_src, e_dst, deg_s, deg_t, E);

    // Layer 1, source->target : h_t = relu(sage(src_emb, tgt_emb, st1))
    hipMemsetAsync(agg, 0, (size_t)NT * H * sizeof(float), stream);
    scatter_add_kernel<<<SCATTER_GRID, 256, 0, stream>>>(src_emb, e_src, e_dst, agg, E);
    sage_gemm_kernel<true><<<(NT + 31) / 32, 256, 0, stream>>>(
        agg, deg_t, tgt_emb, Wl_st1, Wr_st1, b_st1, h_t, NT);

    // Layer 1, target->source : h_s = relu(sage(tgt_emb, src_emb, ts1))
    hipMemsetAsync(agg, 0, (size_t)NS * H * sizeof(float), stream);
    scatter_add_kernel<<<SCATTER_GRID, 256, 0, stream>>>(tgt_emb, e_dst, e_src, agg, E);
    sage_gemm_kernel<true><<<(NS + 31) / 32, 256, 0, stream>>>(
        agg, deg_s, src_emb, Wl_ts1, Wr_ts1, b_ts1, h_s, NS);

    // Layer 2, source->target : o_t = sage(h_s, h_t, st2)
    hipMemsetAsync(agg, 0, (size_t)NT * H * sizeof(float), stream);
    scatter_add_kernel<<<SCATTER_GRID, 256, 0, stream>>>(h_s, e_src, e_dst, agg, E);
    sage_gemm_kernel<false><<<(NT + 31) / 32, 256, 0, stream>>>(
        agg, deg_t, h_t, Wl_st2, Wr_st2, b_st2, o_t, NT);

    // Layer 2, target->source : o_s = sage(h_t, h_s, ts2)
    hipMemsetAsync(agg, 0, (size_t)NS * H * sizeof(float), stream);
    scatter_add_kernel<<<SCATTER_GRID, 256, 0, stream>>>(h_t, e_dst, e_src, agg, E);
    sage_gemm_kernel<false><<<(NS + 31) / 32, 256, 0, stream>>>(
        agg, deg_s, h_s, Wl_ts2, Wr_ts2, b_ts2, o_s, NS);

    // Link classifier.
    dot_kernel<<<2048, 256, 0, stream>>>(o_s, o_t, l_src, l_dst, (float*)d_out, EL);
}